// ApproximationLayer_84327387890499
// MI455X (gfx1250) — compile-verified
//
#include <hip/hip_runtime.h>
#include <stdint.h>

// ---------------------------------------------------------------------------
// Op: out = x, except at (b, rows[i], cols[j]) where bit 62 of the fp64 bit
// pattern is cleared. Pure bandwidth-bound (1 GiB traffic, ~46us @ 23.3TB/s).
// Strategy: bulk async-LDS streaming copy + tiny masked fix-up kernel.
// ---------------------------------------------------------------------------

#if defined(__gfx1250__) && __has_builtin(__builtin_amdgcn_global_load_async_to_lds_b128) && \
    __has_builtin(__builtin_amdgcn_global_store_async_from_lds_b128)
#define USE_ASYNC_LDS 1
#else
#define USE_ASYNC_LDS 0
#endif

// The builtins expect pointers to a 16-byte int vector:
//   arg0: AS(1) (global) v4i*,  arg1: AS(3) (LDS) v4i*
typedef int v4i __attribute__((vector_size(16)));
typedef __attribute__((address_space(1))) v4i gv4i;
typedef __attribute__((address_space(3))) v4i lv4i;

__device__ __forceinline__ void wait_async0() {
#if defined(__gfx1250__)
#if __has_builtin(__builtin_amdgcn_s_wait_asynccnt)
  __builtin_amdgcn_s_wait_asynccnt(0);
#else
  asm volatile("s_wait_asynccnt 0" ::: "memory");
#endif
#endif
}

#define CPY_THREADS 256  // 8 wave32s per block
#define CPY_TILE 4       // 4 x b128 per thread per burst (64B/thread, 16KB LDS/block)

__global__ __launch_bounds__(CPY_THREADS) void stream_copy_kernel(
    const double2* __restrict__ src, double2* __restrict__ dst, size_t n2) {
#if USE_ASYNC_LDS
  __shared__ v4i buf[CPY_THREADS * CPY_TILE];
  const unsigned tid = threadIdx.x;
  const size_t chunk = (size_t)CPY_THREADS * CPY_TILE;
  const size_t stride = chunk * gridDim.x;
  for (size_t base = (size_t)blockIdx.x * chunk; base < n2; base += stride) {
    // Burst of 4 async b128 loads: global -> LDS, ASYNCcnt goes to 4.
#pragma unroll
    for (int k = 0; k < CPY_TILE; ++k) {
      size_t idx = base + (size_t)k * CPY_THREADS + tid;
      if (idx < n2) {
        __builtin_amdgcn_global_load_async_to_lds_b128(
            (gv4i*)(src + idx),
            (lv4i*)(&buf[k * CPY_THREADS + tid]),
            /*offset=*/0, /*cpol=*/0);
      }
    }
    wait_async0();  // loads landed in LDS
    // Burst of 4 async b128 stores: LDS -> global.
#pragma unroll
    for (int k = 0; k < CPY_TILE; ++k) {
      size_t idx = base + (size_t)k * CPY_THREADS + tid;
      if (idx < n2) {
        __builtin_amdgcn_global_store_async_from_lds_b128(
            (gv4i*)(dst + idx),
            (lv4i*)(&buf[k * CPY_THREADS + tid]),
            /*offset=*/0, /*cpol=*/0);
      }
    }
    wait_async0();  // LDS buffer reusable next iteration
  }
#else
  // Fallback: direct VGPR streaming copy with b128 accesses.
  size_t i = (size_t)blockIdx.x * blockDim.x + threadIdx.x;
  const size_t stride = (size_t)gridDim.x * blockDim.x;
  for (; i < n2; i += stride) dst[i] = src[i];
#endif
}

// Fix-up: gather the masked grid from x, clear bit 62, scatter into out.
// Volume: B * nr * nc = 262,144 elements (2 MiB) -> negligible.
__global__ __launch_bounds__(256) void mute_kernel(
    const double* __restrict__ x, const int* __restrict__ rows,
    const int* __restrict__ cols, double* __restrict__ out,
    int R, int C, int nr, int nc, long long total) {
  long long idx = (long long)blockIdx.x * blockDim.x + threadIdx.x;
  if (idx >= total) return;
  int ci = (int)(idx % nc);
  long long t = idx / nc;
  int ri = (int)(t % nr);
  int b = (int)(t / nr);
  int r = rows[ri];
  int c = cols[ci];
  size_t off = ((size_t)b * R + r) * C + c;
  long long bits = __double_as_longlong(x[off]);
  bits &= ~(1ll << 62);  // clear exponent MSB
  out[off] = __longlong_as_double(bits);
}

extern "C" void kernel_launch(void* const* d_in, const int* in_sizes, int n_in,
                              void* d_out, int out_size, void* d_ws, size_t ws_size,
                              hipStream_t stream) {
  const double* x = (const double*)d_in[0];
  const int* rows = (const int*)d_in[1];
  const int* cols = (const int*)d_in[2];
  double* out = (double*)d_out;

  const int R = 512, C = 512;
  const long long total = (long long)in_sizes[0];  // B*R*C (even)
  const int nr = in_sizes[1];
  const int nc = in_sizes[2];
  const int B = (int)(total / ((long long)R * C));

  // 1) Bulk copy as double2 (16B per lane per transfer).
  const size_t n2 = (size_t)total / 2;
  stream_copy_kernel<<<4096, CPY_THREADS, 0, stream>>>(
      (const double2*)x, (double2*)out, n2);

  // 2) Overwrite the masked grid with exponent-MSB-cleared values.
  const long long mtotal = (long long)B * nr * nc;
  const int mblocks = (int)((mtotal + 255) / 256);
  mute_kernel<<<mblocks, 256, 0, stream>>>(x, rows, cols, out, R, C, nr, nc, mtotal);
}